// C3_layer_87419764342871
// MI455X (gfx1250) — compile-verified
//
#include <hip/hip_runtime.h>

typedef float v2f __attribute__((ext_vector_type(2)));
typedef float v8f __attribute__((ext_vector_type(8)));

#define CIN   6
#define COUT  16
#define HIN   256
#define WIN   256
#define HOUTL 252
#define WOUTL 252
#define TY    6                 // 252 = 6 * 42
#define TX    128
#define IN_H  (TY + 4)          // 10 input rows per tile
#define IN_W  (TX + 4)          // 132 input cols per tile
#define NGRP  38                // 30 aligned (dx 0..3) + 8 tail (dx==4) K-groups
#define WLDS  (NGRP * 64)       // weights: [group][half][n(16)][j(2)] = 2432 floats

// LeNet-5 C3 sparse connectivity tables
__device__ const int CH3_tbl[6][3] = {{0,1,2},{1,2,3},{2,3,4},{3,4,5},{0,4,5},{0,1,5}};
__device__ const int CH4_tbl[9][4] = {{0,1,2,3},{1,2,3,4},{2,3,4,5},{0,3,4,5},{0,1,4,5},
                                      {0,1,2,5},{0,1,3,4},{1,2,4,5},{0,2,3,5}};

// LDS offset (in floats) of the dx==4 column for tail slot t (row index t = cin*5+dy).
// Slots t >= 30 are K-padding: weight is zero, so point at a harmless valid address.
__device__ __forceinline__ int tail_off(int t) {
  return (t < 30) ? (((t / 5) * IN_H + (t % 5)) * IN_W + 4) : 0;
}

// Dense-kernel value for output channel n, input channel cin, tap r = dy*5+dx.
__device__ __forceinline__ float dense_w(int n, int cin, int r,
                                         const float* w3, const float* w4,
                                         const float* w6) {
  if (n < 6) {
    #pragma unroll
    for (int j = 0; j < 3; ++j)
      if (CH3_tbl[n][j] == cin) return w3[(n * 3 + j) * 25 + r];
    return 0.0f;
  } else if (n < 15) {
    const int m = n - 6;
    #pragma unroll
    for (int j = 0; j < 4; ++j)
      if (CH4_tbl[m][j] == cin) return w4[(m * 4 + j) * 25 + r];
    return 0.0f;
  }
  return w6[cin * 25 + r];
}

__global__ __launch_bounds__(256, 1) void c3_wmma_f32(
    const float* __restrict__ x,
    const float* __restrict__ w3, const float* __restrict__ b3,
    const float* __restrict__ w4, const float* __restrict__ b4,
    const float* __restrict__ w6, const float* __restrict__ b6,
    float* __restrict__ out) {
  __shared__ float w_lds[WLDS];
  __shared__ float bias_lds[16];
  __shared__ float in_lds[CIN * IN_H * IN_W];   // 7920 floats

  const int tid = threadIdx.x;
  const int b   = blockIdx.z;
  const int y0  = blockIdx.y * TY;
  const int x0  = blockIdx.x ? (WOUTL - TX) : 0;   // 0 or 124 (overlap is benign)

  // --- build dense weights, reordered for WMMA A-fragment b64 loads ---
  // layout: w_lds[g*64 + h*32 + n*2 + j], hardware K-slot s = 2h + j
  for (int idx = tid; idx < WLDS; idx += 256) {
    const int g = idx >> 6, rem = idx & 63;
    const int h = rem >> 5, n = (rem >> 1) & 15, j = rem & 1;
    const int s = 2 * h + j;
    float wv = 0.0f;
    int row = -1, dx = 0;
    if (g < 30) { row = g; dx = s; }                       // dx 0..3 of filter row g
    else { const int t = (g - 30) * 4 + s;                 // dx==4 columns
           if (t < 30) { row = t; dx = 4; } }
    if (row >= 0)
      wv = dense_w(n, row / 5, (row % 5) * 5 + dx, w3, w4, w6);
    w_lds[idx] = wv;
  }
  if (tid < 16)
    bias_lds[tid] = (tid < 6) ? b3[tid] : ((tid < 15) ? b4[tid - 6] : b6[0]);

  // --- stage input tile: 6 cin x 10 rows x 132 cols, float4 (16B-aligned) ---
  for (int idx = tid; idx < CIN * IN_H * (IN_W / 4); idx += 256) {
    const int c  = idx / (IN_H * (IN_W / 4));
    const int rr = idx % (IN_H * (IN_W / 4));
    const int ry = rr / (IN_W / 4);
    const int rx = rr % (IN_W / 4);
    const float4 v =
        ((const float4*)(x + (((size_t)b * CIN + c) * HIN + (y0 + ry)) * WIN + x0))[rx];
    ((float4*)(in_lds + (c * IN_H + ry) * IN_W))[rx] = v;
  }
  __syncthreads();

  const int wave   = tid >> 5;
  const int lane   = tid & 31;
  const int nloc   = lane & 15;    // N: x-offset within wave tile; also channel for A
  const int laneHi = lane >> 4;    // K half-select (slots 0,1 vs 2,3)
  const int xw     = wave * 16;
  const int h2     = 2 * laneHi;

  float bv[8];
  #pragma unroll
  for (int r = 0; r < 8; ++r) bv[r] = bias_lds[r + 8 * laneHi];   // C row M = r + 8*hi

  const float* aw   = w_lds + laneHi * 32 + nloc * 2;   // A: + g*64 (+j)
  const float* binp = in_lds + xw + nloc;               // B: + row offset (+dx)

  #pragma unroll
  for (int yl = 0; yl < TY; ++yl) {
    const float* biny = binp + yl * IN_W;
    v8f acc;
    #pragma unroll
    for (int r = 0; r < 8; ++r) acc[r] = bv[r];

    #pragma unroll
    for (int g = 0; g < NGRP; ++g) {
      // A fragment: two adjacent floats -> ds_load_b64 (shared across fused y-rows)
      const v2f a = *(const v2f*)(aw + g * 64);
      v2f bf;
      if (g < 30) {
        // aligned group: K-slots map to dx = s, same input row -> adjacent cols
        const int ro = ((g / 5) * IN_H + (g % 5)) * IN_W + h2;   // consts fold at unroll
        bf.x = biny[ro];
        bf.y = biny[ro + 1];
      } else {
        // tail group: 4 different filter rows, dx==4
        const int t0 = (g - 30) * 4;
        const int c0 = tail_off(t0 + 0), c1 = tail_off(t0 + 1);
        const int c2 = tail_off(t0 + 2), c3 = tail_off(t0 + 3);
        bf.x = biny[laneHi ? c2 : c0];
        bf.y = biny[laneHi ? c3 : c1];
      }
      // D = A(16ch x 4k) * B(4k x 16x) + C, fp32 exact
      acc = __builtin_amdgcn_wmma_f32_16x16x4_f32(
          false, a, false, bf, (short)0, acc, false, false);
    }

    // C/D layout: vgpr r, lanes0-15 -> (M=r, N=lane); lanes16-31 -> (M=r+8)
    #pragma unroll
    for (int r = 0; r < 8; ++r) {
      const int ch = r + 8 * laneHi;
      out[(((size_t)b * COUT + ch) * HOUTL + (y0 + yl)) * WOUTL + (x0 + xw + nloc)] =
          acc[r];
    }
  }
}

extern "C" void kernel_launch(void* const* d_in, const int* in_sizes, int n_in,
                              void* d_out, int out_size, void* d_ws, size_t ws_size,
                              hipStream_t stream) {
  const float* x  = (const float*)d_in[0];
  const float* w3 = (const float*)d_in[1];
  const float* b3 = (const float*)d_in[2];
  const float* w4 = (const float*)d_in[3];
  const float* b4 = (const float*)d_in[4];
  const float* w6 = (const float*)d_in[5];
  const float* b6 = (const float*)d_in[6];
  float* out = (float*)d_out;
  const int batch = in_sizes[0] / (CIN * HIN * WIN);
  dim3 grid(2, HOUTL / TY, batch);   // x-tiles, y-tiles (42), batch
  c3_wmma_f32<<<grid, dim3(256), 0, stream>>>(x, w3, b3, w4, b4, w6, b6, out);
}